// SNNAvgPool2d_1692217115109
// MI455X (gfx1250) — compile-verified
//
#include <hip/hip_runtime.h>
#include <hip/hip_bf16.h>
#include <stdint.h>

// Problem constants (from the reference)
#define B_   8
#define C_   32
#define H_   64
#define W_   64
#define T_   100
#define OH_  32
#define OW_  32
#define NW_  4      // temporal windows
#define WIN_ 25     // window length
#define ROW_FLOATS   (2 * W_ * T_)        // 12800 floats per block (two H-lines)
#define ROW_FLOAT4   (ROW_FLOATS / 4)     // 3200
#define OUT_ROW_F4   (OW_ * T_ / 4)       // 800

typedef __attribute__((address_space(3))) void as3_void;
typedef float v2f __attribute__((ext_vector_type(2)));
typedef float v8f __attribute__((ext_vector_type(8)));

#if defined(__has_builtin)
#if __has_builtin(__builtin_amdgcn_wmma_f32_16x16x4_f32)
#define HAVE_WMMA_F32X4 1
#endif
#endif

__device__ __forceinline__ float spike_mask(float sum) {
    // Binary spikes => pooled count == pooled sum; keep reference's guarded form.
    float avg = (sum > 0.0f) ? (sum / fmaxf(sum, 1.0f)) : 0.0f;
    return (avg >= 0.125f) ? 1.0f : 0.0f;
}

__global__ __launch_bounds__(256)
void SNNAvgPool2d_1692217115109_kernel(const float* __restrict__ x,
                                       float* __restrict__ out) {
    __shared__ float buf[ROW_FLOATS];   // [h2][w][t] contiguous slice of input
    __shared__ float mask_s[NW_ * OW_]; // indexed by output id = win*32 + ow

    const int tid = threadIdx.x;
    const int bid = blockIdx.x;              // bid = (b*C + c)*OH + oh
    const int oh  = bid & (OH_ - 1);
    const int bc  = bid >> 5;                // b*C + c

    // ---- Phase 1: async copy two contiguous H-lines (51.2 KB) into LDS ----
    const size_t in_off = (size_t)bc * (H_ * W_ * T_) + (size_t)(2 * oh) * (W_ * T_);
    const float4* gsrc = (const float4*)(x + in_off);
    uint32_t lds_base = (uint32_t)(uintptr_t)(as3_void*)buf;

    for (int i = tid; i < ROW_FLOAT4; i += 256) {
        uint32_t lds_addr = lds_base + ((uint32_t)i << 4);
        const float4* g = gsrc + i;
        asm volatile("global_load_async_to_lds_b128 %0, %1, off"
                     :: "v"(lds_addr), "v"(g)
                     : "memory");
    }
    asm volatile("s_wait_asynccnt 0x0" ::: "memory");
    __syncthreads();

    // ---- Phase 2: WMMA reduction. Each wave owns 16 outputs (M dim). ----
    // B = all-ones(4x16)  =>  D[m,n] = sum_k A[m,k] + C[m,n]  (any B striping).
    // A layout (f32 16x4): lane L supplies A[L&15, 2*(L>>4)+e] in element e,
    // i.e. two consecutive t-values per lane.
#if HAVE_WMMA_F32X4
    {
        const int lane    = tid & 31;
        const int wave    = tid >> 5;            // 0..7
        const int m       = lane & 15;
        const int hi      = lane >> 4;           // 0/1 -> K offset 0 / 2
        const int outbase = wave * 16;           // outputs [outbase, outbase+16)
        const int win     = outbase >> 5;        // uniform per wave
        const int ow      = (outbase & 31) + m;  // this lane's output column

        v8f acc = {};
        const v2f bones = {1.0f, 1.0f};

        #pragma unroll
        for (int p = 0; p < 4; ++p) {            // 2x2 spatial window
            const int h2 = p >> 1, dw = p & 1;
            const float* prow = buf + ((h2 * W_) + 2 * ow + dw) * T_ + win * WIN_;
            #pragma unroll
            for (int c = 0; c < 6; ++c) {        // t = 0..23 in chunks of K=4
                v2f a;
                a.x = prow[c * 4 + 2 * hi + 0];
                a.y = prow[c * 4 + 2 * hi + 1];
                acc = __builtin_amdgcn_wmma_f32_16x16x4_f32(
                          false, a, false, bones, (short)0, acc, false, false);
            }
            // remainder t = 24 (K lanes 1..3 masked; load address in-bounds)
            {
                float last = prow[24];
                v2f a; a.x = hi ? 0.0f : last; a.y = 0.0f;
                acc = __builtin_amdgcn_wmma_f32_16x16x4_f32(
                          false, a, false, bones, (short)0, acc, false, false);
            }
        }
        // D layout: column N=0 -> lane 0 has M=0..7 in acc[0..7], lane 16 has M=8..15
        if ((lane & 15) == 0) {
            #pragma unroll
            for (int r = 0; r < 8; ++r)
                mask_s[outbase + hi * 8 + r] = spike_mask(acc[r]);
        }
    }
#else
    if (tid < NW_ * OW_) {
        const int ow  = tid & (OW_ - 1);
        const int win = tid >> 5;
        float sum = 0.0f;
        #pragma unroll
        for (int h2 = 0; h2 < 2; ++h2)
            #pragma unroll
            for (int dw = 0; dw < 2; ++dw) {
                const float* p = buf + ((h2 * W_) + 2 * ow + dw) * T_ + win * WIN_;
                #pragma unroll
                for (int t = 0; t < WIN_; ++t) sum += p[t];
            }
        mask_s[tid] = spike_mask(sum);
    }
#endif
    __syncthreads();

    // ---- Phase 3: write full output row (zeros + scattered masks) ----
    // mask positions t = 12, 37, 62, 87  ->  float4 q = 3(.x), 9(.y), 15(.z), 21(.w)
    float4* orow = (float4*)(out + (size_t)bid * (OW_ * T_));
    for (int i = tid; i < OUT_ROW_F4; i += 256) {
        const int ow = i / (T_ / 4);
        const int q  = i - ow * (T_ / 4);
        float4 v = make_float4(0.0f, 0.0f, 0.0f, 0.0f);
        if (q == 3)  v.x = mask_s[0 * OW_ + ow];
        if (q == 9)  v.y = mask_s[1 * OW_ + ow];
        if (q == 15) v.z = mask_s[2 * OW_ + ow];
        if (q == 21) v.w = mask_s[3 * OW_ + ow];
        orow[i] = v;
    }
}

extern "C" void kernel_launch(void* const* d_in, const int* in_sizes, int n_in,
                              void* d_out, int out_size, void* d_ws, size_t ws_size,
                              hipStream_t stream) {
    (void)in_sizes; (void)n_in; (void)d_ws; (void)ws_size; (void)out_size;
    const float* x = (const float*)d_in[0];
    float* out = (float*)d_out;
    const int nblocks = B_ * C_ * OH_;  // 8192, one per (b,c,oh) output row
    SNNAvgPool2d_1692217115109_kernel<<<nblocks, 256, 0, stream>>>(x, out);
}